// HashGrid_40681930227971
// MI455X (gfx1250) — compile-verified
//
#include <hip/hip_runtime.h>
#include <hip/hip_bf16.h>

typedef __attribute__((ext_vector_type(2))) float v2f;

#define HG_NUM_LEVELS 16
#define HG_F 2

// ---------------- compile-time metadata (replicates Python _build_meta) ----
struct Meta {
  int scales[HG_NUM_LEVELS];       // per-level resolution
  int offs[HG_NUM_LEVELS];         // per-level entry offset (fits in int32)
  int start_hash;
  unsigned long long table;        // first prime >= 2^19
};

constexpr bool kIsPrime(unsigned long long n) {
  if (n < 2) return false;
  for (unsigned long long i = 2; i * i <= n; ++i)
    if (n % i == 0) return false;
  return true;
}

constexpr double kPowInt(double b, int e) {
  double r = 1.0;
  for (int i = 0; i < e; ++i) r *= b;
  return r;
}

// 15th root via Newton (double): matches 128**(1/15) to ~1 ulp; every
// int(16*b^i) sits >= 0.01 from an integer boundary, so truncation matches
// Python's double pow exactly.
constexpr double kRoot15(double v) {
  double x = 1.4;
  for (int it = 0; it < 64; ++it) {
    double x14 = kPowInt(x, 14);
    x = x - (x * x14 - v) / (15.0 * x14);
  }
  return x;
}

constexpr Meta kMakeMeta() {
  Meta m{};
  unsigned long long T = 1ull << 19;
  while (!kIsPrime(T)) ++T;
  const double b = kRoot15(128.0);  // (2048/16)^(1/15)
  long long off = 0;
  int sh = -1;
  for (int i = 0; i < HG_NUM_LEVELS; ++i) {
    int res = (int)(16.0 * kPowInt(b, i));
    m.scales[i] = res;
    m.offs[i] = (int)off;
    long long n = (long long)(res + 1) * (res + 1) * (res + 1);
    if (n > (long long)T) { if (sh < 0) sh = i; n = (long long)T; }
    off += n;
  }
  m.start_hash = (sh < 0) ? HG_NUM_LEVELS : sh;
  m.table = T;
  return m;
}

constexpr Meta kMeta = kMakeMeta();
static_assert(kMeta.start_hash == 5, "expected hashing to start at level 5");
static_assert(kMeta.table >= (1ull << 19) && kMeta.table < (1ull << 20), "table prime sanity");

constexpr unsigned kP32 = (unsigned)kMeta.table;                 // prime, < 2^20
constexpr unsigned kR32 = (unsigned)((1ull << 32) % kMeta.table); // 2^32 mod p
constexpr int kStartHash = kMeta.start_hash;

// Packed per-level metadata; level is wave-uniform -> one scalar b64 load.
struct alignas(8) LevelMeta { int scale; int off; };
struct LvTable { LevelMeta lv[HG_NUM_LEVELS]; };

constexpr LvTable kMakeLv() {
  LvTable t{};
  for (int i = 0; i < HG_NUM_LEVELS; ++i) {
    t.lv[i].scale = kMeta.scales[i];
    t.lv[i].off = kMeta.offs[i];
  }
  return t;
}
__constant__ LvTable g_lv = kMakeLv();

// Exact X mod p for X < 2^38 given split words, using two 32-bit
// constant-modulus reductions (compiler magic-multiply, no u64 divide):
//   X mod p = ((X_lo mod p) + X_hi * (2^32 mod p)) mod p
// Bound: X_hi <= 63, so X_hi*kR32 + (X_lo mod p) < 2^26 + 2^20 fits u32.
__device__ __forceinline__ unsigned mod_p_38(unsigned lo, unsigned hi) {
  unsigned t = (lo % kP32) + hi * kR32;
  return t % kP32;
}

// ---------------- normalize kernel (once per point) -------------------------
// Exact reference pipeline: clamp -> sub -> IEEE f32 div. Writing this once
// per point removes ~40 VALU (3 IEEE divides + clamps) from each of the 16
// per-level threads.
__global__ __launch_bounds__(256) void normalize_pts(
    const float* __restrict__ xyz, const float* __restrict__ wb,
    float* __restrict__ xn, int n_points) {
  const int point = blockIdx.x * blockDim.x + threadIdx.x;
  if (point >= n_points) return;
  const float lo0 = wb[0], lo1 = wb[1], lo2 = wb[2];
  const float hi0 = wb[3], hi1 = wb[4], hi2 = wb[5];
  const float denom = fmaxf(fmaxf(hi0 - lo0, hi1 - lo1), hi2 - lo2) + 1e-6f;
  const float px = xyz[3 * point + 0];
  const float py = xyz[3 * point + 1];
  const float pz = xyz[3 * point + 2];
  xn[3 * point + 0] = (fminf(fmaxf(px, lo0), hi0) - lo0) / denom;
  xn[3 * point + 1] = (fminf(fmaxf(py, lo1), hi1) - lo1) / denom;
  xn[3 * point + 2] = (fminf(fmaxf(pz, lo2), hi2) - lo2) / denom;
}

// ---------------- main kernel ----------------------------------------------
// One thread per (point, level); level = blockIdx.y is wave-uniform, so the
// dense/hash split is a scalar branch and level metadata lives in SGPRs.
// Lanes = consecutive points -> coalesced coordinate loads. Output lines are
// filled by the 16 level-waves and merge in the 192MB L2 before writeback.
template <bool PRE>
__global__ __launch_bounds__(256) void hashgrid_fwd(
    const float* __restrict__ pos,   // PRE ? normalized coords : raw xyz
    const float* __restrict__ wb,
    const float* __restrict__ data,
    float* __restrict__ out,
    int n_points) {
  const int point = blockIdx.x * blockDim.x + threadIdx.x;
  const int level = blockIdx.y;
  if (point >= n_points) return;

  float x0, x1, x2;
  if (PRE) {
    x0 = pos[3 * point + 0];
    x1 = pos[3 * point + 1];
    x2 = pos[3 * point + 2];
  } else {
    const float lo0 = wb[0], lo1 = wb[1], lo2 = wb[2];
    const float hi0 = wb[3], hi1 = wb[4], hi2 = wb[5];
    const float denom = fmaxf(fmaxf(hi0 - lo0, hi1 - lo1), hi2 - lo2) + 1e-6f;
    x0 = (fminf(fmaxf(pos[3 * point + 0], lo0), hi0) - lo0) / denom;
    x1 = (fminf(fmaxf(pos[3 * point + 1], lo1), hi1) - lo1) / denom;
    x2 = (fminf(fmaxf(pos[3 * point + 2], lo2), hi2) - lo2) / denom;
  }

  const LevelMeta lm = g_lv.lv[level];   // uniform -> scalar load
  const int res = lm.scale;
  const unsigned lvl_off = (unsigned)lm.off;
  const float s = (float)res;

  const float fx0 = x0 * s, fx1 = x1 * s, fx2 = x2 * s;

  // JAX: int_xyz = float32(fx + corner) truncated (fx >= 0); do the +1
  // corners through the same f32 add to match edge rounding exactly.
  const unsigned ilo0 = (unsigned)(int)fx0;
  const unsigned ilo1 = (unsigned)(int)fx1;
  const unsigned ilo2 = (unsigned)(int)fx2;
  const unsigned ihi0 = (unsigned)(int)(fx0 + 1.0f);
  const unsigned ihi1 = (unsigned)(int)(fx1 + 1.0f);
  const unsigned ihi2 = (unsigned)(int)(fx2 + 1.0f);

  // off = fx - trunc(fx): exact; v_fract_f32 computes it in one op (fx >= 0).
  const float off0 = __builtin_amdgcn_fractf(fx0);
  const float off1 = __builtin_amdgcn_fractf(fx1);
  const float off2 = __builtin_amdgcn_fractf(fx2);

  unsigned i0, i1, i2, i3, i4, i5, i6, i7;
  if (level < kStartHash) {
    // dense levels (res <= 59): pure 32-bit strided index
    const unsigned sp1 = (unsigned)(res + 1);
    const unsigned sp1sq = sp1 * sp1;          // scalar math
    const unsigned dX0 = ilo0 * sp1sq + lvl_off;
    const unsigned dX1 = ihi0 * sp1sq + lvl_off;
    const unsigned dY0 = ilo1 * sp1;
    const unsigned dY1 = ihi1 * sp1;
    i0 = dX0 + dY0 + ilo2;
    i1 = dX0 + dY0 + ihi2;
    i2 = dX0 + dY1 + ilo2;
    i3 = dX0 + dY1 + ihi2;
    i4 = dX1 + dY0 + ilo2;
    i5 = dX1 + dY0 + ihi2;
    i6 = dX1 + dY1 + ilo2;
    i7 = dX1 + dY1 + ihi2;
  } else {
    // hashed levels: products < 2^38 as (lo,hi) word pairs, hoisted per dim
    const unsigned yL0 = ilo1 * 19349663u, yH0 = __umulhi(ilo1, 19349663u);
    const unsigned yL1 = ihi1 * 19349663u, yH1 = __umulhi(ihi1, 19349663u);
    const unsigned zL0 = ilo2 * 83492791u, zH0 = __umulhi(ilo2, 83492791u);
    const unsigned zL1 = ihi2 * 83492791u, zH1 = __umulhi(ihi2, 83492791u);
    i0 = mod_p_38(ilo0 ^ yL0 ^ zL0, yH0 ^ zH0) + lvl_off;
    i1 = mod_p_38(ilo0 ^ yL0 ^ zL1, yH0 ^ zH1) + lvl_off;
    i2 = mod_p_38(ilo0 ^ yL1 ^ zL0, yH1 ^ zH0) + lvl_off;
    i3 = mod_p_38(ilo0 ^ yL1 ^ zL1, yH1 ^ zH1) + lvl_off;
    i4 = mod_p_38(ihi0 ^ yL0 ^ zL0, yH0 ^ zH0) + lvl_off;
    i5 = mod_p_38(ihi0 ^ yL0 ^ zL1, yH0 ^ zH1) + lvl_off;
    i6 = mod_p_38(ihi0 ^ yL1 ^ zL0, yH1 ^ zH0) + lvl_off;
    i7 = mod_p_38(ihi0 ^ yL1 ^ zL1, yH1 ^ zH1) + lvl_off;
  }

  // All 8 gathers issued before the weight math; 32-bit element index against
  // SGPR base -> GVS addressing with scale_offset, 8 loads in flight.
  const v2f v0 = *(const v2f*)(data + 2ull * i0);
  const v2f v1 = *(const v2f*)(data + 2ull * i1);
  const v2f v2 = *(const v2f*)(data + 2ull * i2);
  const v2f v3 = *(const v2f*)(data + 2ull * i3);
  const v2f v4 = *(const v2f*)(data + 2ull * i4);
  const v2f v5 = *(const v2f*)(data + 2ull * i5);
  const v2f v6 = *(const v2f*)(data + 2ull * i6);
  const v2f v7 = *(const v2f*)(data + 2ull * i7);

  // trilinear weights; reference's clip to [0,1] is a provable no-op
  // (off in [0,1), 1-off in (0,1], RN stays in range) and is elided.
  const float wl0 = 1.0f - off0;
  const float wl1 = 1.0f - off1;
  const float wl2 = 1.0f - off2;

  const float w0 = wl0 * wl1 * wl2;
  const float w1 = wl0 * wl1 * off2;
  const float w2 = wl0 * off1 * wl2;
  const float w3 = wl0 * off1 * off2;
  const float w4 = off0 * wl1 * wl2;
  const float w5 = off0 * wl1 * off2;
  const float w6 = off0 * off1 * wl2;
  const float w7 = off0 * off1 * off2;

  float a0 = w0 * v0.x + w1 * v1.x + w2 * v2.x + w3 * v3.x
           + w4 * v4.x + w5 * v5.x + w6 * v6.x + w7 * v7.x;
  float a1 = w0 * v0.y + w1 * v1.y + w2 * v2.y + w3 * v3.y
           + w4 * v4.y + w5 * v5.y + w6 * v6.y + w7 * v7.y;

  // Output [N, L*F]: element offset point*32 + level*2. Regular (RT) store:
  // the 16 level-waves' 8B pieces of each 128B line merge in L2 (output 16MB
  // + table 49MB << 192MB L2) before a single writeback.
  v2f r;
  r.x = a0;
  r.y = a1;
  *((v2f*)out + (long long)point * HG_NUM_LEVELS + level) = r;
}

// ---------------- launch ----------------------------------------------------
extern "C" void kernel_launch(void* const* d_in, const int* in_sizes, int n_in,
                              void* d_out, int out_size, void* d_ws, size_t ws_size,
                              hipStream_t stream) {
  const float* xyz  = (const float*)d_in[0];   // [N,3] f32
  const float* wb   = (const float*)d_in[1];   // [6]   f32
  const float* data = (const float*)d_in[2];   // [OFFSETS[-1], 2] f32
  float* out = (float*)d_out;                  // [N, 32] f32

  const int n_points = in_sizes[0] / 3;
  const int block = 256;
  const int gx = (n_points + block - 1) / block;
  const dim3 grid(gx, HG_NUM_LEVELS);

  const size_t need_ws = (size_t)n_points * 3 * sizeof(float);
  if (ws_size >= need_ws) {
    float* xn = (float*)d_ws;
    hipLaunchKernelGGL(normalize_pts, dim3(gx), dim3(block), 0, stream,
                       xyz, wb, xn, n_points);
    hipLaunchKernelGGL((hashgrid_fwd<true>), grid, dim3(block), 0, stream,
                       xn, wb, data, out, n_points);
  } else {
    hipLaunchKernelGGL((hashgrid_fwd<false>), grid, dim3(block), 0, stream,
                       xyz, wb, data, out, n_points);
  }
}